// ContextGateAttention_56195352100980
// MI455X (gfx1250) — compile-verified
//
#include <hip/hip_runtime.h>
#include <hip/hip_bf16.h>

typedef __attribute__((ext_vector_type(2))) float v2f;
typedef __attribute__((ext_vector_type(4))) float v4f;
typedef __attribute__((ext_vector_type(8))) float v8f;

#define B_  4
#define L_  512
#define D_  128
#define BL_ (B_ * L_)
#define SCALE 0.08838834764831845f  /* 1/sqrt(128) */

// ---------------------------------------------------------------------------
// Kernel 1: hqb = Q @ W1[:D] + b1 ; hk = K @ W1[D:]
// X = [Q; K] rows 0..2*BL-1. One WG = 16 rows; wave w = output cols [16w,16w+16).
// V_WMMA_F32_16X16X4_F32 accumulating over the 128-deep contraction.
// ---------------------------------------------------------------------------
__global__ __launch_bounds__(256)
void cga_proj_kernel(const float* __restrict__ Q, const float* __restrict__ K,
                     const float* __restrict__ W1, const float* __restrict__ b1,
                     float* __restrict__ hqb, float* __restrict__ hk)
{
    const int r0   = blockIdx.x * 16;          // row tile in X (0..4095)
    const bool isK = (r0 >= BL_);
    const float* X = isK ? (K + (size_t)(r0 - BL_) * D_) : (Q + (size_t)r0 * D_);

    const int tid  = threadIdx.x;
    const int wave = tid >> 5;
    const int lane = tid & 31;
    const int half = lane >> 4;        // K-pair selector for A/B operands
    const int lp   = lane & 15;        // m (A) / n (B,C,D)

    __shared__ float ldsX[16][132];    // +4 pad: stride-132 kills bank conflicts

    { // cooperative 16x128 tile stage: 256 threads x 8 floats
        const int row = tid >> 4;
        const int col = (tid & 15) * 8;
        const float* src = X + (size_t)row * D_ + col;
        *(float4*)&ldsX[row][col]     = *(const float4*)(src);
        *(float4*)&ldsX[row][col + 4] = *(const float4*)(src + 4);
    }
    __syncthreads();

    const int d0   = wave * 16;        // output column tile
    const int woff = isK ? D_ : 0;     // which half of W1 [256,128]

    v8f acc = {};
    for (int c = 0; c < D_; c += 4) {
        // A: X[m][c + kb + j], contiguous float2 per lane
        v2f a = *(const v2f*)&ldsX[lp][c + 2 * half];
        // B: W1[woff + c + kb + j][d0 + n] — two coalesced scalar reads
        const float* wr = W1 + (size_t)(woff + c + 2 * half) * D_ + d0 + lp;
        v2f bv; bv.x = wr[0]; bv.y = wr[D_];
        acc = __builtin_amdgcn_wmma_f32_16x16x4_f32(false, a, false, bv,
                                                    (short)0, acc, false, false);
    }

    float* dst      = isK ? hk : hqb;
    const int rbase = isK ? (r0 - BL_) : r0;
#pragma unroll
    for (int r = 0; r < 8; ++r) {
        const int row = rbase + r + 8 * half;      // C/D layout: M = r + 8*half
        float v = acc[r];
        if (!isK) v += b1[d0 + lp];                // fold b1 into hqb
        dst[(size_t)row * D_ + d0 + lp] = v;
    }
}

// ---------------------------------------------------------------------------
// Kernel 2: fused gated attention for one (b, 16-row q tile).
// 8 waves: each owns 4 k-tiles (S via WMMA + conflict via VALU + gate),
// then per-row wave32 softmax, then P@V via WMMA (one d-tile per wave).
// ---------------------------------------------------------------------------
__global__ __launch_bounds__(256)
void cga_attn_kernel(const float* __restrict__ Q, const float* __restrict__ K,
                     const float* __restrict__ V, const int* __restrict__ mask,
                     const float* __restrict__ W2, const float* __restrict__ b2,
                     const float* __restrict__ hqb, const float* __restrict__ hk,
                     float* __restrict__ out, float* __restrict__ weights)
{
    const int b  = blockIdx.x >> 5;    // batch
    const int q0 = (blockIdx.x & 31) * 16;

    const int tid  = threadIdx.x;
    const int wave = tid >> 5;
    const int lane = tid & 31;
    const int half = lane >> 4;
    const int lp   = lane & 15;

    __shared__ float ldsQ[16][132];    // Q tile (padded)
    __shared__ float ldsH[16][132];    // hqb tile (b1 folded)
    __shared__ float ldsW2[128];
    __shared__ float SG[16][516];      // gated scores, +4 pad for P@V column reads

    { // stage Q tile + hqb tile + W2
        const int row = tid >> 4;
        const int col = (tid & 15) * 8;
        const float* qs = Q   + ((size_t)(b * L_) + q0 + row) * D_ + col;
        const float* hs = hqb + ((size_t)(b * L_) + q0 + row) * D_ + col;
        *(float4*)&ldsQ[row][col]     = *(const float4*)(qs);
        *(float4*)&ldsQ[row][col + 4] = *(const float4*)(qs + 4);
        *(float4*)&ldsH[row][col]     = *(const float4*)(hs);
        *(float4*)&ldsH[row][col + 4] = *(const float4*)(hs + 4);
        if (tid < 128) ldsW2[tid] = W2[tid];
    }
    __syncthreads();

    const float bias2 = b2[0];

    for (int kt = wave; kt < 32; kt += 8) {
        const int k0 = kt * 16;

        // ---- S = Q_tile * K_tileᵀ  (32 chained f32 WMMAs) ----
        v8f acc = {};
        const float* krow = K + ((size_t)(b * L_) + k0 + lp) * D_ + 2 * half;
        for (int c = 0; c < D_; c += 4) {
            v2f a  = *(const v2f*)&ldsQ[lp][c + 2 * half];
            v2f bv = *(const v2f*)(krow + c);   // B[kb+j][n] == K[k0+n][c+kb+j]
            acc = __builtin_amdgcn_wmma_f32_16x16x4_f32(false, a, false, bv,
                                                        (short)0, acc, false, false);
        }

        // ---- conflict[q,k] = Σ_d relu(hqb[q,d]+hk[k,d])·W2[d]  (VALU) ----
        float conf[8] = {0.f, 0.f, 0.f, 0.f, 0.f, 0.f, 0.f, 0.f};
        const float* hkr = hk + ((size_t)(b * L_) + k0 + lp) * D_;
        for (int d = 0; d < D_; d += 4) {
            v4f hv = *(const v4f*)(hkr + d);        // coalesced global float4
            v4f wv = *(const v4f*)&ldsW2[d];        // LDS broadcast
#pragma unroll
            for (int r = 0; r < 8; ++r) {
                v4f hq = *(const v4f*)&ldsH[r + 8 * half][d];  // LDS broadcast
                conf[r] += fmaxf(hq.x + hv.x, 0.f) * wv.x
                         + fmaxf(hq.y + hv.y, 0.f) * wv.y
                         + fmaxf(hq.z + hv.z, 0.f) * wv.z
                         + fmaxf(hq.w + hv.w, 0.f) * wv.w;
            }
        }

        // ---- gate + mask -> SG ----
#pragma unroll
        for (int r = 0; r < 8; ++r) {
            const int qrow = r + 8 * half;
            const float g  = 1.f / (1.f + __expf(-(conf[r] + bias2)));
            float sg = acc[r] * SCALE * g;
            const int mv = mask[(((size_t)b * L_) + q0 + qrow) * L_ + k0 + lp];
            if (mv == 0) sg = -__builtin_inff();
            SG[qrow][k0 + lp] = sg;
        }
    }
    __syncthreads();

    // ---- softmax over k (wave32 shuffle reduction; 2 rows per wave) ----
    for (int rr = 0; rr < 2; ++rr) {
        const int row = wave * 2 + rr;
        float mx = -__builtin_inff();
        for (int c = lane; c < L_; c += 32) mx = fmaxf(mx, SG[row][c]);
        for (int off = 16; off; off >>= 1) mx = fmaxf(mx, __shfl_xor(mx, off, 32));
        float sum = 0.f;
        for (int c = lane; c < L_; c += 32) {
            const float e = __expf(SG[row][c] - mx);
            SG[row][c] = e;
            sum += e;
        }
        for (int off = 16; off; off >>= 1) sum += __shfl_xor(sum, off, 32);
        const float inv = 1.f / sum;
        float* wout = weights + (((size_t)b * L_) + q0 + row) * L_;
        for (int c = lane; c < L_; c += 32) {
            const float p = SG[row][c] * inv;
            SG[row][c] = p;
            wout[c] = p;
        }
    }
    __syncthreads();

    // ---- out = P @ V  (128 chained f32 WMMAs; wave owns d-tile = 16*wave) ----
    {
        const int d0 = wave * 16;
        v8f acc = {};
        for (int k4 = 0; k4 < L_; k4 += 4) {
            v2f a = *(const v2f*)&SG[lp][k4 + 2 * half];     // A: P[m][k4+kb+j]
            const float* vr = V + ((size_t)(b * L_) + k4 + 2 * half) * D_ + d0 + lp;
            v2f bv; bv.x = vr[0]; bv.y = vr[D_];             // B: V[k4+kb+j][d0+n]
            acc = __builtin_amdgcn_wmma_f32_16x16x4_f32(false, a, false, bv,
                                                        (short)0, acc, false, false);
        }
#pragma unroll
        for (int r = 0; r < 8; ++r)
            out[(((size_t)b * L_) + q0 + r + 8 * half) * D_ + d0 + lp] = acc[r];
    }
}

// ---------------------------------------------------------------------------
extern "C" void kernel_launch(void* const* d_in, const int* in_sizes, int n_in,
                              void* d_out, int out_size, void* d_ws, size_t ws_size,
                              hipStream_t stream)
{
    const float* Q    = (const float*)d_in[0];
    const float* K    = (const float*)d_in[1];
    const float* V    = (const float*)d_in[2];
    const int*   mask = (const int*)  d_in[3];
    const float* W1   = (const float*)d_in[4];
    const float* b1   = (const float*)d_in[5];
    const float* W2   = (const float*)d_in[6];
    const float* b2   = (const float*)d_in[7];

    float* out     = (float*)d_out;                       // [B,L,D]
    float* weights = (float*)d_out + (size_t)BL_ * D_;    // [B,L,L]

    // workspace: hqb [B,L,D] then hk [B,L,D]  (2 MB total)
    float* hqb = (float*)d_ws;
    float* hk  = (float*)d_ws + (size_t)BL_ * D_;

    cga_proj_kernel<<<2 * BL_ / 16, 256, 0, stream>>>(Q, K, W1, b1, hqb, hk);
    cga_attn_kernel<<<B_ * (L_ / 16), 256, 0, stream>>>(Q, K, V, mask, W2, b2,
                                                        hqb, hk, out, weights);
}